// MultiHeadAttention_74036646248995
// MI455X (gfx1250) — compile-verified
//
#include <hip/hip_runtime.h>
#include <math.h>

// Problem constants (from reference): B=2, S=2048, DMODEL=1024, HEADS=16, DH=64
#define B_  2
#define S_  2048
#define D_  1024
#define H_  16
#define DH_ 64

typedef __attribute__((ext_vector_type(2))) float v2f;
typedef __attribute__((ext_vector_type(8))) float v8f;
typedef __attribute__((ext_vector_type(4))) unsigned int u32x4;
typedef __attribute__((ext_vector_type(8))) int i32x8;
typedef __attribute__((ext_vector_type(4))) int i32x4;

__device__ __forceinline__ v8f wmma_f32_4(v2f a, v2f b, v8f c) {
    // V_WMMA_F32_16X16X4_F32 : D = A(16x4) * B(4x16) + C(16x16), all fp32
    return __builtin_amdgcn_wmma_f32_16x16x4_f32(
        false, a, false, b, (short)0, c, false, false);
}

// -------------------------------------------------------------------------
// Tensor Data Mover descriptors: 2D fp32 tile (tile1 rows x tile0 cols),
// row-major global tensor with row stride stride0 elements, packed row-major
// in LDS at lds_off. D# per CDNA5 ISA ch.8; groups 2/3 zero => <=2D tensor.
// -------------------------------------------------------------------------
__device__ __forceinline__ void tdm_make_desc(
    u32x4& g0, i32x8& g1,
    unsigned lds_off, const void* gptr,
    unsigned tdim0, unsigned tdim1, unsigned stride0,
    unsigned tile0, unsigned tile1)
{
    const unsigned long long ga = (unsigned long long)(uintptr_t)gptr;
    g0[0] = 1u;                                       // count=1, user D#
    g0[1] = lds_off;                                  // lds_addr (bytes)
    g0[2] = (unsigned)(ga & 0xFFFFFFFFu);             // global_addr[31:0]
    g0[3] = (unsigned)((ga >> 32) & 0x01FFFFFFu)      // global_addr[56:32]
          | (2u << 30);                               // type=2 ("image")

    g1[0] = (int)(2u << 16);                          // data_size=2 -> 4 bytes
    g1[1] = (int)((tdim0 & 0xFFFFu) << 16);           // tensor_dim0[15:0]
    g1[2] = (int)(((tdim0 >> 16) & 0xFFFFu)           // tensor_dim0[31:16]
          |        ((tdim1 & 0xFFFFu) << 16));        // tensor_dim1[15:0]
    g1[3] = (int)(((tdim1 >> 16) & 0xFFFFu)           // tensor_dim1[31:16]
          |        ((tile0 & 0xFFFFu) << 16));        // tile_dim0
    g1[4] = (int)(tile1 & 0xFFFFu);                   // tile_dim1 (tile_dim2=0)
    g1[5] = (int)stride0;                             // tensor_dim0_stride[31:0]
    g1[6] = 0;
    g1[7] = 0;
}

__device__ __forceinline__ void tdm_load_tile_f32(
    unsigned lds_off, const void* gptr,
    unsigned tdim0, unsigned tdim1, unsigned stride0,
    unsigned tile0, unsigned tile1)
{
    u32x4 g0; i32x8 g1;
    tdm_make_desc(g0, g1, lds_off, gptr, tdim0, tdim1, stride0, tile0, tile1);
    const i32x4 z4 = {0, 0, 0, 0};
#if __clang_major__ >= 23
    const i32x8 z8 = {0, 0, 0, 0, 0, 0, 0, 0};
    __builtin_amdgcn_tensor_load_to_lds(g0, g1, z4, z4, z8, 0);
#else
    __builtin_amdgcn_tensor_load_to_lds(g0, g1, z4, z4, 0);
#endif
}

__device__ __forceinline__ void tdm_store_tile_f32(
    unsigned lds_off, void* gptr,
    unsigned tdim0, unsigned tdim1, unsigned stride0,
    unsigned tile0, unsigned tile1)
{
    u32x4 g0; i32x8 g1;
    tdm_make_desc(g0, g1, lds_off, gptr, tdim0, tdim1, stride0, tile0, tile1);
    const i32x4 z4 = {0, 0, 0, 0};
#if __clang_major__ >= 23
    const i32x8 z8 = {0, 0, 0, 0, 0, 0, 0, 0};
    __builtin_amdgcn_tensor_store_from_lds(g0, g1, z4, z4, z8, 0);
#else
    __builtin_amdgcn_tensor_store_from_lds(g0, g1, z4, z4, 0);
#endif
}

__device__ __forceinline__ unsigned lds_addr_of(const void* p) {
    // Generic LDS pointers carry the LDS byte offset in the low 32 bits.
    return (unsigned)(uintptr_t)p;
}

// -------------------------------------------------------------------------
// Kernel 1: per-head flash attention.
// grid = (S/64, B*H), block = 128 threads (4 wave32).
// Each head's Q/K/V is a contiguous (S x DH) fp32 block at bh*S*DH
// (reference uses direct reshape (B,H,S,DH)).
// K/V tiles stream through double-buffered LDS via the Tensor Data Mover:
// wave 0 waits TENSORcnt for tile kb, barrier publishes it, then wave 0
// immediately issues tile kb+1 into the opposite buffer (which the same
// barrier proved free) so the DMA runs under the WMMA compute.
// The 64x64 output tile is staged to LDS and written by one TDM store.
// -------------------------------------------------------------------------
__global__ __launch_bounds__(128) void attn_fwd(
    const float* __restrict__ Q, const float* __restrict__ K,
    const float* __restrict__ V, const float* __restrict__ mask,
    float* __restrict__ X)
{
    __shared__ __align__(16) float Ks[2][64 * DH_];
    __shared__ __align__(16) float Vs[2][64 * DH_];
    __shared__ __align__(16) float Ps[4][16 * 64];   // per-wave P; reused as O tile

    const int tid  = threadIdx.x;
    const int wave = tid >> 5;
    const int lane = tid & 31;
    const int half = lane >> 4;     // 0: lanes 0-15, 1: lanes 16-31
    const int l16  = lane & 15;

    const int bh = blockIdx.y;                       // 0..B*H-1
    const size_t headOff = (size_t)bh * S_ * DH_;
    const float* Qh = Q + headOff;
    const float* Kh = K + headOff;
    const float* Vh = V + headOff;
    float*       Xh = X + headOff;

    const int q0 = blockIdx.x * 64;       // block's query base
    const int qw = q0 + wave * 16;        // this wave's 16 query rows

    // Prologue: kick off tile 0 DMA before anything else.
    if (wave == 0) {
        tdm_load_tile_f32(lds_addr_of(Ks[0]), Kh, DH_, S_, DH_, 64, 64);
        tdm_load_tile_f32(lds_addr_of(Vs[0]), Vh, DH_, S_, DH_, 64, 64);
    }

    // ---- Q fragments in WMMA A-layout, kept in registers for whole kernel.
    v2f qa[16];
    {
        const float* qrow = Qh + (size_t)(qw + l16) * DH_;
        #pragma unroll
        for (int s = 0; s < 16; ++s)
            qa[s] = *reinterpret_cast<const v2f*>(qrow + 4 * s + 2 * half);
    }

    const v8f vzero = {0.f,0.f,0.f,0.f,0.f,0.f,0.f,0.f};
    v8f o[4];
    #pragma unroll
    for (int t = 0; t < 4; ++t) o[t] = vzero;
    float mrow[8], lrow[8];
    #pragma unroll
    for (int r = 0; r < 8; ++r) { mrow[r] = -INFINITY; lrow[r] = 0.f; }

    const float scale = 0.125f;           // 1/sqrt(64)

    for (int kb = 0; kb < S_ / 64; ++kb) {
        const int cur = kb & 1;
        if (wave == 0)
            __builtin_amdgcn_s_wait_tensorcnt(0);   // tile kb resident
        __syncthreads();   // publish tile kb; all waves done with buffer cur^1
        if (wave == 0 && kb + 1 < S_ / 64) {
            tdm_load_tile_f32(lds_addr_of(Ks[cur ^ 1]),
                              Kh + (size_t)(kb + 1) * 64 * DH_,
                              DH_, S_, DH_, 64, 64);
            tdm_load_tile_f32(lds_addr_of(Vs[cur ^ 1]),
                              Vh + (size_t)(kb + 1) * 64 * DH_,
                              DH_, S_, DH_, 64, 64);
        }
        const float* Kc = Ks[cur];
        const float* Vc = Vs[cur];

        // ---- scores: S = Q(16x64) * K^T  -> four 16x16 tiles, K-dim = 64
        v8f sc[4];
        #pragma unroll
        for (int t = 0; t < 4; ++t) sc[t] = vzero;
        #pragma unroll
        for (int s = 0; s < 16; ++s) {
            v2f a = qa[s];
            #pragma unroll
            for (int t = 0; t < 4; ++t) {
                v2f b = *reinterpret_cast<const v2f*>(
                    &Kc[(16 * t + l16) * DH_ + 4 * s + 2 * half]);
                sc[t] = wmma_f32_4(a, b, sc[t]);
            }
        }

        // scale + additive mask (mask shape (1,1,S,S), broadcast over b,h)
        {
            const float* mp = mask + (size_t)(qw + 8 * half) * S_
                                   + (size_t)kb * 64 + l16;
            #pragma unroll
            for (int t = 0; t < 4; ++t)
                #pragma unroll
                for (int r = 0; r < 8; ++r)
                    sc[t][r] = sc[t][r] * scale + mp[(size_t)r * S_ + 16 * t];
        }

        // ---- online softmax (rows in v8f elements; cols across 16-lane half)
        #pragma unroll
        for (int r = 0; r < 8; ++r) {
            float v = fmaxf(fmaxf(sc[0][r], sc[1][r]), fmaxf(sc[2][r], sc[3][r]));
            v = fmaxf(v, __shfl_xor(v, 1, 32));
            v = fmaxf(v, __shfl_xor(v, 2, 32));
            v = fmaxf(v, __shfl_xor(v, 4, 32));
            v = fmaxf(v, __shfl_xor(v, 8, 32));
            const float mn    = fmaxf(mrow[r], v);
            const float alpha = expf(mrow[r] - mn);
            float rs = 0.f;
            #pragma unroll
            for (int t = 0; t < 4; ++t) {
                float p = expf(sc[t][r] - mn);
                sc[t][r] = p;
                rs += p;
            }
            rs += __shfl_xor(rs, 1, 32);
            rs += __shfl_xor(rs, 2, 32);
            rs += __shfl_xor(rs, 4, 32);
            rs += __shfl_xor(rs, 8, 32);
            lrow[r] = lrow[r] * alpha + rs;
            mrow[r] = mn;
            #pragma unroll
            for (int t = 0; t < 4; ++t) o[t][r] *= alpha;
        }

        // ---- C-layout -> A-layout for P via wave-private LDS (in-order DS)
        float* ps = Ps[wave];
        #pragma unroll
        for (int t = 0; t < 4; ++t)
            #pragma unroll
            for (int r = 0; r < 8; ++r)
                ps[(r + 8 * half) * 64 + 16 * t + l16] = sc[t][r];

        // ---- O += P(16x64) * V(64x64), four 16x16 output tiles
        #pragma unroll
        for (int s = 0; s < 16; ++s) {
            v2f a = *reinterpret_cast<const v2f*>(&ps[l16 * 64 + 4 * s + 2 * half]);
            #pragma unroll
            for (int t = 0; t < 4; ++t) {
                v2f b;
                b.x = Vc[(4 * s + 2 * half + 0) * DH_ + 16 * t + l16];
                b.y = Vc[(4 * s + 2 * half + 1) * DH_ + 16 * t + l16];
                o[t] = wmma_f32_4(a, b, o[t]);
            }
        }
    }

    // ---- epilogue: normalize, stage the 64x64 O tile in LDS (reusing Ps,
    // which is exactly 64x64: wave w owns rows 16w..16w+15), TDM-store it.
    float* osh = &Ps[0][0];
    #pragma unroll
    for (int r = 0; r < 8; ++r) {
        const float inv = 1.0f / lrow[r];
        #pragma unroll
        for (int t = 0; t < 4; ++t)
            osh[(wave * 16 + r + 8 * half) * 64 + 16 * t + l16] = o[t][r] * inv;
    }
    __syncthreads();
    if (wave == 0) {
        tdm_store_tile_f32(lds_addr_of(osh), Xh + (size_t)q0 * DH_,
                           DH_, S_, DH_, 64, 64);
        __builtin_amdgcn_s_wait_tensorcnt(0);
    }
}

// -------------------------------------------------------------------------
// Kernel 2: out = X @ W^T + bias.  X:(4096x1024), W:(1024x1024) row-major.
// grid = (N/64, M/64), block = 128 threads (4 waves), each wave does 16x64.
// X/W tiles double-buffered through LDS via TDM; result tile staged to LDS
// and written back with one strided TDM store.
// -------------------------------------------------------------------------
__global__ __launch_bounds__(128) void out_proj(
    const float* __restrict__ X, const float* __restrict__ W,
    const float* __restrict__ bias, float* __restrict__ out)
{
    __shared__ __align__(16) float Xs[2][64 * 64];
    __shared__ __align__(16) float Wsh[2][64 * 64];  // rows indexed by n, cols by k

    const int tid  = threadIdx.x;
    const int wave = tid >> 5;
    const int lane = tid & 31;
    const int half = lane >> 4;
    const int l16  = lane & 15;

    const int n0 = blockIdx.x * 64;
    const int m0 = blockIdx.y * 64;

    if (wave == 0) {
        tdm_load_tile_f32(lds_addr_of(Xs[0]), &X[(size_t)m0 * D_],
                          D_, B_ * S_, D_, 64, 64);
        tdm_load_tile_f32(lds_addr_of(Wsh[0]), &W[(size_t)n0 * D_],
                          D_, D_, D_, 64, 64);
    }

    const v8f vzero = {0.f,0.f,0.f,0.f,0.f,0.f,0.f,0.f};
    v8f acc[4];
    #pragma unroll
    for (int t = 0; t < 4; ++t) acc[t] = vzero;

    for (int kb = 0; kb < D_ / 64; ++kb) {
        const int cur = kb & 1;
        if (wave == 0)
            __builtin_amdgcn_s_wait_tensorcnt(0);
        __syncthreads();
        if (wave == 0 && kb + 1 < D_ / 64) {
            tdm_load_tile_f32(lds_addr_of(Xs[cur ^ 1]),
                              &X[(size_t)m0 * D_ + (kb + 1) * 64],
                              D_, B_ * S_, D_, 64, 64);
            tdm_load_tile_f32(lds_addr_of(Wsh[cur ^ 1]),
                              &W[(size_t)n0 * D_ + (kb + 1) * 64],
                              D_, D_, D_, 64, 64);
        }
        const float* Xc = Xs[cur];
        const float* Wc = Wsh[cur];

        #pragma unroll
        for (int s = 0; s < 16; ++s) {
            v2f a = *reinterpret_cast<const v2f*>(
                &Xc[(wave * 16 + l16) * 64 + 4 * s + 2 * half]);
            #pragma unroll
            for (int t = 0; t < 4; ++t) {
                v2f b = *reinterpret_cast<const v2f*>(
                    &Wc[(16 * t + l16) * 64 + 4 * s + 2 * half]);
                acc[t] = wmma_f32_4(a, b, acc[t]);
            }
        }
    }

    // ---- epilogue: bias add, stage 64x64 tile in LDS (buffer 0 is free:
    // the last compute iteration used buffer 1), strided TDM store to out.
    float* osh = &Xs[0][0];
    #pragma unroll
    for (int t = 0; t < 4; ++t) {
        const float bv = bias[n0 + 16 * t + l16];
        #pragma unroll
        for (int r = 0; r < 8; ++r)
            osh[(wave * 16 + r + 8 * half) * 64 + 16 * t + l16] = acc[t][r] + bv;
    }
    __syncthreads();
    if (wave == 0) {
        tdm_store_tile_f32(lds_addr_of(osh), &out[(size_t)m0 * D_ + n0],
                           D_, B_ * S_, D_, 64, 64);
        __builtin_amdgcn_s_wait_tensorcnt(0);
    }
}

// -------------------------------------------------------------------------
extern "C" void kernel_launch(void* const* d_in, const int* in_sizes, int n_in,
                              void* d_out, int out_size, void* d_ws, size_t ws_size,
                              hipStream_t stream) {
    (void)in_sizes; (void)n_in; (void)out_size; (void)ws_size;
    const float* Q    = (const float*)d_in[0];
    const float* K    = (const float*)d_in[1];
    const float* V    = (const float*)d_in[2];
    const float* mask = (const float*)d_in[3];
    const float* W    = (const float*)d_in[4];
    const float* bias = (const float*)d_in[5];
    float* out = (float*)d_out;
    float* X   = (float*)d_ws;    // B*S*DMODEL fp32 = 16 MB attention output

    dim3 g1(S_ / 64, B_ * H_);            // (32, 32)
    attn_fwd<<<g1, 128, 0, stream>>>(Q, K, V, mask, X);

    dim3 g2(D_ / 64, (B_ * S_) / 64);     // (16, 64)
    out_proj<<<g2, 128, 0, stream>>>(X, W, bias, out);
}